// PairwiseHuberLoss_36593121362282
// MI455X (gfx1250) — compile-verified
//
#include <hip/hip_runtime.h>

#define BDIM 256
#define ITEMS 8
#define TILE 2048
#define SOFT 0.5f
#define HARD 0.1f

typedef float v2f __attribute__((ext_vector_type(2)));
typedef float v8f __attribute__((ext_vector_type(8)));

__device__ __forceinline__ unsigned umin_u(unsigned a, unsigned b){ return a<b?a:b; }

__device__ __forceinline__ unsigned f2key(float f){
  unsigned u = __float_as_uint(f);
  unsigned m = (unsigned)(((int)u) >> 31) | 0x80000000u;
  return u ^ m;
}
__device__ __forceinline__ float key2f(unsigned k){
  unsigned m = (k & 0x80000000u) ? 0x80000000u : 0xFFFFFFFFu;
  return __uint_as_float(k ^ m);
}

// ---- block helpers (deterministic, fixed-order) ----
__device__ unsigned blockScanExclU(unsigned v, unsigned* lds, unsigned* totalOut){
  unsigned t = threadIdx.x;
  lds[t] = v; __syncthreads();
  for (int off = 1; off < BDIM; off <<= 1){
    unsigned x = (t >= (unsigned)off) ? lds[t-off] : 0u;
    __syncthreads();
    lds[t] += x;
    __syncthreads();
  }
  unsigned incl = lds[t];
  unsigned tot  = lds[BDIM-1];
  __syncthreads();
  if (totalOut) *totalOut = tot;
  return incl - v;
}

__device__ float blockReduceF(float v, float* lds){
  unsigned t = threadIdx.x;
  lds[t] = v; __syncthreads();
  for (int off = BDIM/2; off > 0; off >>= 1){
    if (t < (unsigned)off) lds[t] += lds[t+off];
    __syncthreads();
  }
  float r = lds[0];
  __syncthreads();
  return r;
}

// ---- K0: zero counters ----
__global__ void kInit(unsigned* ctr){
  if (threadIdx.x < 64) ctr[threadIdx.x] = 0u;
}

// ---- K1: partition positives to (key,weight), per-block weight sums ----
__global__ void __launch_bounds__(BDIM)
kPartition(const int* labels, const float* scores, const float* weights, int n,
           unsigned* ctr, unsigned* pingK, unsigned* pingV,
           float* posWp, float* negWp){
  __shared__ unsigned sScan[BDIM];
  __shared__ float    sRed[BDIM];
  __shared__ unsigned sBase;
  unsigned t = threadIdx.x;
  size_t base = (size_t)blockIdx.x * TILE;
  unsigned keys[ITEMS], vals[ITEMS];
  unsigned cnt = 0;
  float pw = 0.f, nw = 0.f;
  for (int j = 0; j < ITEMS; j++){
    size_t g = base + t + (size_t)j*BDIM;
    if (g < (size_t)n){
      int lab = labels[g];
      float s = scores[g];
      float w = weights[g];
      if (lab > 0){ keys[cnt]=f2key(s); vals[cnt]=__float_as_uint(w); cnt++; pw += w; }
      else nw += w;
    }
  }
  unsigned tot;
  unsigned exc = blockScanExclU(cnt, sScan, &tot);
  if (t == 0) sBase = atomicAdd(&ctr[0], tot);
  __syncthreads();
  unsigned dst = sBase + exc;
  for (unsigned j = 0; j < cnt; j++){ pingK[dst+j]=keys[j]; pingV[dst+j]=vals[j]; }
  float pws = blockReduceF(pw, sRed);
  float nws = blockReduceF(nw, sRed);
  if (t == 0){ posWp[blockIdx.x]=pws; negWp[blockIdx.x]=nws; }
}

// ---- radix sort: histogram pass ----
__global__ void __launch_bounds__(BDIM)
kHist(const unsigned* inK, const unsigned* ctr, unsigned shift,
      unsigned* hist, int nTiles){
  __shared__ unsigned h[256];
  unsigned t = threadIdx.x;
  h[t] = 0u; __syncthreads();
  unsigned P = ctr[0];
  size_t base = (size_t)blockIdx.x * TILE;
  for (int j = 0; j < ITEMS; j++){
    size_t g = base + t + (size_t)j*BDIM;
    if (g < (size_t)P) atomicAdd(&h[(inK[g] >> shift) & 0xFFu], 1u);
  }
  __syncthreads();
  hist[(size_t)t * nTiles + blockIdx.x] = h[t];
}

// ---- single-block exclusive scan of u32 array (digit-major hist) ----
__global__ void __launch_bounds__(BDIM)
kScanU32(unsigned* a, unsigned long long count){
  __shared__ unsigned sScan[BDIM];
  __shared__ unsigned sCarry;
  unsigned t = threadIdx.x;
  if (t == 0) sCarry = 0u;
  __syncthreads();
  const size_t CH = (size_t)BDIM * 16;
  for (size_t c0 = 0; c0 < count; c0 += CH){
    unsigned loc[16]; unsigned cnt = 0;
    for (int j = 0; j < 16; j++){
      size_t i = c0 + (size_t)t*16 + j;
      loc[j] = (i < count) ? a[i] : 0u;
      cnt += loc[j];
    }
    unsigned tot;
    unsigned exc = blockScanExclU(cnt, sScan, &tot);
    unsigned run = sCarry + exc;
    for (int j = 0; j < 16; j++){
      size_t i = c0 + (size_t)t*16 + j;
      unsigned v = loc[j];
      if (i < count) a[i] = run;
      run += v;
    }
    __syncthreads();
    if (t == 0) sCarry += tot;
    __syncthreads();
  }
}

// ---- radix sort: stable scatter via in-LDS 1-bit splits ----
__global__ void __launch_bounds__(BDIM)
kScatter(const unsigned* inK, const unsigned* inV, unsigned* outK, unsigned* outV,
         const unsigned* ctr, const unsigned* hist, unsigned shift, int nTiles){
  __shared__ unsigned A0[TILE], B0[TILE], A1[TILE], B1[TILE];
  __shared__ unsigned sScan[BDIM];
  __shared__ unsigned dStart[256];
  unsigned t = threadIdx.x;
  unsigned P = ctr[0];
  size_t base = (size_t)blockIdx.x * TILE;
  unsigned valid = (P > base) ? umin_u((unsigned)TILE, (unsigned)(P - base)) : 0u;
  for (int j = 0; j < ITEMS; j++){
    unsigned o = t + j*BDIM;
    size_t g = base + o;
    A0[o] = (o < valid) ? inK[g] : 0xFFFFFFFFu;
    B0[o] = (o < valid) ? inV[g] : 0u;
  }
  __syncthreads();
  unsigned* k0 = A0; unsigned* v0 = B0; unsigned* k1 = A1; unsigned* v1 = B1;
  for (int bit = 0; bit < 8; bit++){
    unsigned kk[ITEMS], vv[ITEMS], bb[ITEMS]; unsigned zc = 0;
    for (int j = 0; j < ITEMS; j++){
      unsigned o = t*ITEMS + j;
      kk[j] = k0[o]; vv[j] = v0[o];
      bb[j] = (kk[j] >> (shift + bit)) & 1u;
      zc += (bb[j] ^ 1u);
    }
    unsigned totZ;
    unsigned zb = blockScanExclU(zc, sScan, &totZ);
    unsigned ob = totZ + (t*ITEMS - zb);
    for (int j = 0; j < ITEMS; j++){
      unsigned d = bb[j] ? ob++ : zb++;
      k1[d] = kk[j]; v1[d] = vv[j];
    }
    __syncthreads();
    unsigned* tk=k0; k0=k1; k1=tk;
    unsigned* tv=v0; v0=v1; v1=tv;
  }
  dStart[t] = 0u; __syncthreads();
  for (int j = 0; j < ITEMS; j++){
    unsigned o = t + j*BDIM;
    unsigned d = (k0[o] >> shift) & 0xFFu;
    if (o == 0 || (((k0[o-1] >> shift) & 0xFFu) != d)) dStart[d] = o;
  }
  __syncthreads();
  for (int j = 0; j < ITEMS; j++){
    unsigned o = t + j*BDIM;
    if (o < valid){
      unsigned key = k0[o];
      unsigned d = (key >> shift) & 0xFFu;
      unsigned gd = hist[(size_t)d * nTiles + blockIdx.x] + (o - dStart[d]);
      outK[gd] = key; outV[gd] = v0[o];
    }
  }
}

// ---- per-block moment totals ----
__global__ void __launch_bounds__(BDIM)
kMomSums(const unsigned* sK, const unsigned* sV, const unsigned* ctr,
         float* bs, int nTiles){
  __shared__ float sRed[BDIM];
  unsigned t = threadIdx.x;
  unsigned P = ctr[0];
  size_t base = (size_t)blockIdx.x * TILE;
  float m0=0.f, m1=0.f, m2=0.f;
  for (int j = 0; j < ITEMS; j++){
    size_t g = base + t + (size_t)j*BDIM;
    if (g < (size_t)P){
      float s = key2f(sK[g]);
      float w = __uint_as_float(sV[g]);
      m0 += w; m1 += w*s; m2 += w*s*s;
    }
  }
  float r0 = blockReduceF(m0, sRed);
  float r1 = blockReduceF(m1, sRed);
  float r2 = blockReduceF(m2, sRed);
  if (t == 0){
    bs[blockIdx.x] = r0;
    bs[(size_t)nTiles + blockIdx.x] = r1;
    bs[(size_t)2*nTiles + blockIdx.x] = r2;
  }
}

// ---- single-block exclusive scan of f32 array ----
__global__ void __launch_bounds__(BDIM)
kScanF32(float* a, int count){
  __shared__ float sS[BDIM];
  __shared__ float sCarry;
  unsigned t = threadIdx.x;
  if (t == 0) sCarry = 0.f;
  __syncthreads();
  const int CH = BDIM * 16;
  for (int c0 = 0; c0 < count; c0 += CH){
    float loc[16]; float cnt = 0.f;
    for (int j = 0; j < 16; j++){
      int i = c0 + (int)t*16 + j;
      loc[j] = (i < count) ? a[i] : 0.f;
      cnt += loc[j];
    }
    sS[t] = cnt; __syncthreads();
    for (int off = 1; off < BDIM; off <<= 1){
      float x = (t >= (unsigned)off) ? sS[t-off] : 0.f;
      __syncthreads();
      sS[t] += x;
      __syncthreads();
    }
    float incl = sS[t]; float tot = sS[BDIM-1];
    __syncthreads();
    float run = sCarry + incl - cnt;
    for (int j = 0; j < 16; j++){
      int i = c0 + (int)t*16 + j;
      float v = loc[j];
      if (i < count) a[i] = run;
      run += v;
    }
    __syncthreads();
    if (t == 0) sCarry += tot;
    __syncthreads();
  }
}

// ---- WMMA-based local prefix scan of the three moments ----
// Per wave: 256 elements arranged as X[k][n] (16 pos x 16 seg), scanned by
// D = L * X chained over 4 K-slices of V_WMMA_F32_16X16X4_F32 (L = lower-tri ones).
// Moment loop is force-unrolled so each moment gets a branch-free 4xWMMA chain.
__global__ void __launch_bounds__(BDIM)
kScanWMMA(const unsigned* sK, const unsigned* sV, const unsigned* ctr,
          const float* bs, int nTiles,
          float* S0, float* S1, float* S2){
  __shared__ float segTot[8][16];
  __shared__ float waveTot[8];
  unsigned t = threadIdx.x;
  unsigned lane = t & 31u;
  unsigned wv = t >> 5;
  unsigned P = ctr[0];
  size_t waveBase = (size_t)blockIdx.x * TILE + (size_t)wv * 256;
  unsigned seg = lane & 15u;
  unsigned hi  = (lane >> 4) & 1u;  // 0: lanes 0-15 hold K 0,1 (+4s); 1: lanes 16-31 hold K 2,3 (+4s)
  unsigned M = lane & 15u;

  // load this lane's 8 (seg,pos) elements straight into WMMA B-slice ownership,
  // and expand to the three moment value sets up-front (no branches near WMMA).
  float bv[3][8];
  #pragma unroll
  for (int s4 = 0; s4 < 4; s4++){
    #pragma unroll
    for (int e = 0; e < 2; e++){
      unsigned p = 4u*s4 + 2u*hi + (unsigned)e;
      size_t g = waveBase + (size_t)seg*16 + p;
      bool ok = g < (size_t)P;
      float s = ok ? key2f(sK[g]) : 0.f;
      float w = ok ? __uint_as_float(sV[g]) : 0.f;
      int i = s4*2 + e;
      bv[0][i] = w;
      bv[1][i] = w * s;
      bv[2][i] = w * s * s;
    }
  }

  // constant A slices of the lower-triangular ones matrix (per-lane constants)
  float ax[4], ay[4];
  #pragma unroll
  for (int s4 = 0; s4 < 4; s4++){
    unsigned k0 = 4u*(unsigned)s4 + 2u*hi;
    ax[s4] = (k0      <= M) ? 1.f : 0.f;
    ay[s4] = (k0 + 1u <= M) ? 1.f : 0.f;
  }

  #pragma unroll
  for (int m = 0; m < 3; m++){
    v8f acc = {};
    #pragma unroll
    for (int s4 = 0; s4 < 4; s4++){
      v2f a, b;
      a.x = ax[s4]; a.y = ay[s4];
      b.x = bv[m][s4*2 + 0];
      b.y = bv[m][s4*2 + 1];
      acc = __builtin_amdgcn_wmma_f32_16x16x4_f32(false, a, false, b,
                                                  (short)0, acc, false, false);
    }
    // acc[v] = inclusive scan at position (v + 8*hi) of segment `seg`
    if (hi) segTot[wv][seg] = acc[7];   // position 15 = segment total
    __syncthreads();
    float cross = 0.f;
    for (unsigned nseg = 0; nseg < seg; nseg++) cross += segTot[wv][nseg];
    float wtot = 0.f;
    for (unsigned nseg = 0; nseg < 16; nseg++) wtot += segTot[wv][nseg];
    if (lane == 0) waveTot[wv] = wtot;
    __syncthreads();
    float woff = bs[(size_t)m * nTiles + blockIdx.x];   // exclusive block offset
    for (unsigned w2 = 0; w2 < wv; w2++) woff += waveTot[w2];
    float* Sm = (m == 0) ? S0 : ((m == 1) ? S1 : S2);
    #pragma unroll
    for (int v = 0; v < 8; v++){
      unsigned p = (unsigned)v + 8u*hi;
      size_t g = waveBase + (size_t)seg*16 + p;
      if (g < (size_t)P) Sm[g+1] = acc[v] + cross + woff;
    }
    __syncthreads();
  }
  if (blockIdx.x == 0 && t == 0){ S0[0]=0.f; S1[0]=0.f; S2[0]=0.f; }
}

// ---- fenced lower_bound: 11 LDS probes + ~11 L2 probes ----
__device__ unsigned lowerBoundFenced(const unsigned* keys, const unsigned* fence,
                                     unsigned numFence, unsigned step, unsigned P, unsigned q){
  unsigned lo = 0, hiF = numFence;
  while (lo < hiF){ unsigned mid = (lo + hiF) >> 1; if (fence[mid] < q) lo = mid + 1; else hiF = mid; }
  unsigned kf = lo;
  unsigned glo = kf ? (kf - 1u) * step : 0u;
  unsigned ghi = (kf < numFence) ? umin_u(kf * step + 1u, P) : P;
  while (glo < ghi){ unsigned mid = (glo + ghi) >> 1; if (keys[mid] < q) glo = mid + 1; else ghi = mid; }
  return glo;
}

// ---- loss over negatives (fence staged into LDS via async-to-LDS) ----
__global__ void __launch_bounds__(BDIM)
kLoss(const int* labels, const float* scores, const float* weights, int n,
      const unsigned* sK, const unsigned* ctr,
      const float* S0, const float* S1, const float* S2,
      float* lossP){
  __shared__ unsigned fence[2048];
  __shared__ float sRed[BDIM];
  unsigned t = threadIdx.x;
  unsigned P = ctr[0];
  unsigned step = (P + 2047u) / 2048u; if (step == 0u) step = 1u;
  unsigned numFence = P ? ((P + step - 1u) / step) : 0u;

  if (P){
    for (int j = 0; j < 8; j++){
      unsigned k = t + (unsigned)j*BDIM;
      unsigned gi = umin_u(k * step, P - 1u);
      const unsigned* gp = sK + gi;
      unsigned ldsOff = (unsigned)(size_t)&fence[k];
      // CDNA5 async copy: global -> LDS, tracked by ASYNCcnt
      asm volatile("global_load_async_to_lds_b32 %0, %1, off"
                   :: "v"(ldsOff), "v"(gp) : "memory");
    }
    asm volatile("s_wait_asynccnt 0" ::: "memory");
  }
  __syncthreads();

  size_t base = (size_t)blockIdx.x * TILE;
  float acc = 0.f;
  const float inv2s = 1.0f / (2.0f * SOFT);
  for (int j = 0; j < ITEMS; j++){
    size_t g = base + t + (size_t)j*BDIM;
    if (g < (size_t)n){
      int lab = labels[g];
      if (lab <= 0 && P > 0u){
        float ns = scores[g];
        float wn = weights[g];
        float blo = ns + (HARD - SOFT*0.5f);
        float bhi = ns + (HARD + SOFT*0.5f);
        unsigned ql = f2key(blo), qh = f2key(bhi);
        unsigned bl = lowerBoundFenced(sK, fence, numFence, step, P, ql);
        unsigned bh = lowerBoundFenced(sK, fence, numFence, step, P, qh);
        float p0l = S0[bl], p1l = S1[bl], p2l = S2[bl];
        float p0h = S0[bh], p1h = S1[bh], p2h = S2[bh];
        float lin = p0l * (ns + HARD) - p1l;
        float tt = ns + HARD + SOFT*0.5f;
        float quad = (p0h - p0l) * (tt*tt*inv2s)
                   - (p1h - p1l) * (tt / SOFT)
                   + (p2h - p2l) * inv2s;
        acc += (lin + quad) * wn;
      }
    }
  }
  float r = blockReduceF(acc, sRed);
  if (t == 0) lossP[blockIdx.x] = r;
}

// ---- final deterministic reduction + normalization ----
__global__ void __launch_bounds__(BDIM)
kFinal(const float* lossP, const float* posWp, const float* negWp,
       int nTiles, float* out){
  __shared__ float sRed[BDIM];
  unsigned t = threadIdx.x;
  float a=0.f, b=0.f, c=0.f;
  for (int i = (int)t; i < nTiles; i += BDIM){ a += lossP[i]; b += posWp[i]; c += negWp[i]; }
  float L  = blockReduceF(a, sRed);
  float PW = blockReduceF(b, sRed);
  float NW = blockReduceF(c, sRed);
  if (t == 0) out[0] = L / (PW * NW);
}

extern "C" void kernel_launch(void* const* d_in, const int* in_sizes, int n_in,
                              void* d_out, int out_size, void* d_ws, size_t ws_size,
                              hipStream_t stream) {
  (void)n_in; (void)out_size; (void)ws_size;
  const int*   labels  = (const int*)  d_in[0];
  const float* scores  = (const float*)d_in[1];
  const float* weights = (const float*)d_in[2];
  float* out = (float*)d_out;
  int n = in_sizes[0];
  int nTiles = (n + TILE - 1) / TILE;
  size_t NP = (size_t)n + 16;

  char* ws = (char*)d_ws;
  auto alignUp = [](size_t x){ return (x + 255) & ~(size_t)255; };
  size_t off = 0;
  unsigned* ctr   = (unsigned*)(ws + off); off = alignUp(off + 256);
  unsigned* pingK = (unsigned*)(ws + off); off = alignUp(off + NP*4);
  unsigned* pingV = (unsigned*)(ws + off); off = alignUp(off + NP*4);
  unsigned* pongK = (unsigned*)(ws + off); off = alignUp(off + NP*4);
  unsigned* pongV = (unsigned*)(ws + off); off = alignUp(off + NP*4);
  float*    S2    = (float*)   (ws + off); off = alignUp(off + NP*4);
  unsigned* hist  = (unsigned*)(ws + off); off = alignUp(off + (size_t)256*nTiles*4);
  float*    bs    = (float*)   (ws + off); off = alignUp(off + (size_t)3*nTiles*4);
  float*    posWp = (float*)   (ws + off); off = alignUp(off + (size_t)nTiles*4);
  float*    negWp = (float*)   (ws + off); off = alignUp(off + (size_t)nTiles*4);
  float*    lossP = (float*)   (ws + off); off = alignUp(off + (size_t)nTiles*4);
  // S0/S1 alias the pong buffers (free after the last sort pass lands in ping)
  float* S0 = (float*)pongK;
  float* S1 = (float*)pongV;

  kInit<<<1, 64, 0, stream>>>(ctr);
  kPartition<<<nTiles, BDIM, 0, stream>>>(labels, scores, weights, n,
                                          ctr, pingK, pingV, posWp, negWp);

  unsigned* ik = pingK; unsigned* iv = pingV;
  unsigned* ok = pongK; unsigned* ov = pongV;
  for (int p = 0; p < 4; p++){
    kHist   <<<nTiles, BDIM, 0, stream>>>(ik, ctr, (unsigned)(p*8), hist, nTiles);
    kScanU32<<<1,      BDIM, 0, stream>>>(hist, (unsigned long long)256 * (unsigned long long)nTiles);
    kScatter<<<nTiles, BDIM, 0, stream>>>(ik, iv, ok, ov, ctr, hist, (unsigned)(p*8), nTiles);
    unsigned* tk = ik; ik = ok; ok = tk;
    unsigned* tv = iv; iv = ov; ov = tv;
  }
  // after 4 passes sorted data is back in pingK/pingV

  kMomSums<<<nTiles, BDIM, 0, stream>>>(pingK, pingV, ctr, bs, nTiles);
  kScanF32<<<1, BDIM, 0, stream>>>(bs,                       nTiles);
  kScanF32<<<1, BDIM, 0, stream>>>(bs + (size_t)nTiles,      nTiles);
  kScanF32<<<1, BDIM, 0, stream>>>(bs + (size_t)2*nTiles,    nTiles);
  kScanWMMA<<<nTiles, BDIM, 0, stream>>>(pingK, pingV, ctr, bs, nTiles, S0, S1, S2);

  kLoss<<<nTiles, BDIM, 0, stream>>>(labels, scores, weights, n,
                                     pingK, ctr, S0, S1, S2, lossP);
  kFinal<<<1, BDIM, 0, stream>>>(lossP, posWp, negWp, nTiles, out);
}